// SSIM_68324339745171
// MI455X (gfx1250) — compile-verified
//
#include <hip/hip_runtime.h>

typedef __attribute__((ext_vector_type(2))) float v2f;
typedef __attribute__((ext_vector_type(4))) float v4f;
typedef __attribute__((ext_vector_type(8))) float v8f;

#define SSIM_C1 (0.01f * 0.01f)
#define SSIM_C2 (0.03f * 0.03f)

// Column stride (in floats) of the transpose bounce buffer: keeps 16B alignment for
// b128 stores (80*col + 32*hk bytes) and 20x mod 64 is injective over x in [0,16).
#define VSTRIDE 20

static __device__ __forceinline__ v8f wmma4(v2f a, v2f b, v8f c) {
    // D = A(16x4,f32) * B(4x16,f32) + C(16x16,f32)
    return __builtin_amdgcn_wmma_f32_16x16x4_f32(false, a, false, b, (short)0, c, false, false);
}

static __device__ __forceinline__ v8f v8f_zero() {
    v8f z;
#pragma unroll
    for (int i = 0; i < 8; ++i) z[i] = 0.0f;
    return z;
}

__global__ void ssim_zero(double* acc) { *acc = 0.0; }

// One wave computes one 16x16 output tile.
//   Step A (vertical blur):   V(16x32) = Gv(16x28) x In(28x32)   -> 2x7 WMMA chains x5 quantities
//   Step B (horizontal blur): D(16x16) = V(16x28) x Gh(28x16)    -> 7 WMMA chain x5 quantities
// Gv[m][r] = g[r-m], Gh[r][x] = g[r-x]; per-lane fragments of both are identical.
template <bool BORDER>
static __device__ __forceinline__ float tile_ssim_sum(const float* __restrict__ p1,
                                                      const float* __restrict__ p2,
                                                      float* __restrict__ vb,
                                                      const v2f gfrag[7],
                                                      int y0, int x0, int xl, int hk)
{
    // ---------------- Step A: vertical blur for all 5 quantities ----------------
    v8f accQ[5][2];
#pragma unroll
    for (int q = 0; q < 5; ++q)
#pragma unroll
        for (int n = 0; n < 2; ++n) accQ[q][n] = v8f_zero();

    // Per-lane base address for the interior fast path: element (row y0-5+2hk, col x0-5+xl)
    const int baseOff = (y0 - 5 + 2 * hk) * 512 + (x0 - 5 + xl);
    const float* b1p = p1 + baseOff;
    const float* b2p = p2 + baseOff;

#pragma unroll
    for (int n = 0; n < 2; ++n) {
        const int xg = x0 - 5 + 16 * n + xl;
        const int xc = min(max(xg, 0), 511);
        const bool xok = ((unsigned)xg) < 512u;
#pragma unroll
        for (int kc = 0; kc < 7; ++kc) {
            v2f b1, b2;
#pragma unroll
            for (int j = 0; j < 2; ++j) {
                if (BORDER) {
                    const int yg = y0 - 5 + 2 * hk + 4 * kc + j;
                    const int yc = min(max(yg, 0), 511);
                    const bool ok = xok && (((unsigned)yg) < 512u);
                    const int o = yc * 512 + xc;           // always in-bounds
                    const float v1 = p1[o];
                    const float v2 = p2[o];
                    b1[j] = ok ? v1 : 0.0f;                // zero 'SAME' padding
                    b2[j] = ok ? v2 : 0.0f;
                } else {
                    const int o = (4 * kc + j) * 512 + 16 * n;   // compile-time immediate
                    b1[j] = b1p[o];
                    b2[j] = b2p[o];
                }
            }
            const v2f b11 = b1 * b1;
            const v2f b22 = b2 * b2;
            const v2f b12 = b1 * b2;
            accQ[0][n] = wmma4(gfrag[kc], b1,  accQ[0][n]);
            accQ[1][n] = wmma4(gfrag[kc], b2,  accQ[1][n]);
            accQ[2][n] = wmma4(gfrag[kc], b11, accQ[2][n]);
            accQ[3][n] = wmma4(gfrag[kc], b22, accQ[3][n]);
            accQ[4][n] = wmma4(gfrag[kc], b12, accQ[4][n]);
        }
    }

    // ---------------- Step B: horizontal blur via LDS layout bounce ----------------
    // Column-major bounce: vbT[N][M] with column stride VSTRIDE floats.
    // D-layout lane holds (M = r + 8*hk, N = 16n + xl) for r=0..7 -> 8 contiguous floats
    // per (q,n) => two ds_store_b128. Same-wave DS ops are in-order, no barrier needed.
    v8f d[5];
#pragma unroll
    for (int q = 0; q < 5; ++q) {
#pragma unroll
        for (int n = 0; n < 2; ++n) {
            float* dst = vb + (16 * n + xl) * VSTRIDE + 8 * hk;
            v4f lo, hi;
#pragma unroll
            for (int r = 0; r < 4; ++r) { lo[r] = accQ[q][n][r]; hi[r] = accQ[q][n][r + 4]; }
            *(v4f*)(dst)     = lo;
            *(v4f*)(dst + 4) = hi;
        }

        // A-layout fragments: lane needs V[m=xl][k=4kc+2hk+j] = vbT[k][m]
        v8f accB = v8f_zero();
#pragma unroll
        for (int kc = 0; kc < 7; ++kc) {
            v2f a;
            a[0] = vb[(4 * kc + 2 * hk + 0) * VSTRIDE + xl];
            a[1] = vb[(4 * kc + 2 * hk + 1) * VSTRIDE + xl];
            accB = wmma4(a, gfrag[kc], accB);
        }
        d[q] = accB;
    }

    // ---------------- Pointwise SSIM map ----------------
    float lsum = 0.0f;
#pragma unroll
    for (int r = 0; r < 8; ++r) {
        const float mu1 = d[0][r], mu2 = d[1][r];
        const float e11 = d[2][r], e22 = d[3][r], e12 = d[4][r];
        const float mu1sq = mu1 * mu1;
        const float mu2sq = mu2 * mu2;
        const float mu12  = mu1 * mu2;
        const float s1  = e11 - mu1sq;
        const float s2  = e22 - mu2sq;
        const float s12 = e12 - mu12;
        const float num = (2.0f * mu12 + SSIM_C1) * (2.0f * s12 + SSIM_C2);
        const float den = (mu1sq + mu2sq + SSIM_C1) * (s1 + s2 + SSIM_C2);
        lsum += num / den;
    }
    return lsum;
}

__global__ __launch_bounds__(256) void ssim_main(const float* __restrict__ img1,
                                                 const float* __restrict__ img2,
                                                 double* __restrict__ acc)
{
    __shared__ float gsh[11];
    __shared__ __align__(16) float vbuf[8][32 * VSTRIDE];  // per-wave col-major bounce

    const int tid = threadIdx.x;
    if (tid == 0) {
        double w[11];
        double s = 0.0;
#pragma unroll
        for (int t = 0; t < 11; ++t) { double d = (double)(t - 5); w[t] = exp(-(d * d) / 4.5); s += w[t]; }
#pragma unroll
        for (int t = 0; t < 11; ++t) gsh[t] = (float)(w[t] / s);
    }
    __syncthreads();

    const int wave = tid >> 5;
    const int lane = tid & 31;
    const int xl   = lane & 15;   // row (A layout) or column (B/C layouts)
    const int hk   = lane >> 4;   // lane-half selects K pair / M+8

    const int tileId = blockIdx.x * 8 + wave;     // 0..49151
    const int imgIdx = tileId >> 10;              // 48 images of 512x512
    const int rem    = tileId & 1023;
    const int ty = rem >> 5;
    const int tx = rem & 31;
    const int y0 = ty << 4;
    const int x0 = tx << 4;

    const float* p1 = img1 + (size_t)imgIdx * (512u * 512u);
    const float* p2 = img2 + (size_t)imgIdx * (512u * 512u);

    // Band-matrix fragments (identical for Gv A-side and Gh B-side), branch-free gather.
    v2f gfrag[7];
#pragma unroll
    for (int kc = 0; kc < 7; ++kc) {
        v2f a;
#pragma unroll
        for (int j = 0; j < 2; ++j) {
            const int idx = 4 * kc + 2 * hk + j - xl;
            const int idc = min(max(idx, 0), 10);
            const float gv = gsh[idc];
            a[j] = (idx == idc) ? gv : 0.0f;
        }
        gfrag[kc] = a;
    }

    float* vb = &vbuf[wave][0];

    // Wave-uniform interior/border selection -> scalar branch (EXEC stays all-ones for WMMA).
    const int borderV = ((tx == 0) | (tx == 31) | (ty == 0) | (ty == 31)) ? 1 : 0;
    const int border  = __builtin_amdgcn_readfirstlane(borderV);

    float lsum;
    if (border)
        lsum = tile_ssim_sum<true >(p1, p2, vb, gfrag, y0, x0, xl, hk);
    else
        lsum = tile_ssim_sum<false>(p1, p2, vb, gfrag, y0, x0, xl, hk);

#pragma unroll
    for (int off = 16; off > 0; off >>= 1)
        lsum += __shfl_down(lsum, off, 32);
    if (lane == 0)
        atomicAdd(acc, (double)lsum);
}

__global__ void ssim_final(const double* __restrict__ acc, float* __restrict__ out) {
    const double n = 16.0 * 3.0 * 512.0 * 512.0;
    out[0] = 1.0f - (float)(acc[0] / n);
}

extern "C" void kernel_launch(void* const* d_in, const int* in_sizes, int n_in,
                              void* d_out, int out_size, void* d_ws, size_t ws_size,
                              hipStream_t stream) {
    const float* img1 = (const float*)d_in[0];
    const float* img2 = (const float*)d_in[1];
    double* acc = (double*)d_ws;

    ssim_zero<<<1, 1, 0, stream>>>(acc);
    // 49152 tiles of 16x16 (48 images x 32x32 tiles), 8 waves/block, 1 tile/wave
    ssim_main<<<6144, 256, 0, stream>>>(img1, img2, acc);
    ssim_final<<<1, 1, 0, stream>>>(acc, (float*)d_out);
}